// BoundaryLoss_8624294331222
// MI455X (gfx1250) — compile-verified
//
#include <hip/hip_runtime.h>
#include <hip/hip_bf16.h>

// CDNA5 (gfx1250) boundary-loss kernel.
//
// Key insight: target is binary, so min(EDT(m==0), EDT(m!=0)) == 0 at every
// pixel (each pixel is a member of one of the two zero-sets). Hence
// w == 1 everywhere and final_weight == 1 + THETA == 6 exactly.
// Loss = (6/N) * sum( softplus(pred) - pred*target ).
// This removes the O(H^2*W) EDT entirely; what remains is a 4 MB streaming
// reduction (~0.18 us at 23.3 TB/s) -> bandwidth bound.

typedef float v2f __attribute__((ext_vector_type(2)));
typedef float v8f __attribute__((ext_vector_type(8)));

__device__ __forceinline__ float softplus_bce(float p, float t) {
    // logaddexp(0, p) - p*t, numerically stable (matches jnp.logaddexp)
    float sp = fmaxf(p, 0.0f) + log1pf(expf(-fabsf(p)));
    return __builtin_fmaf(-p, t, sp);
}

// Full 32-lane wave reduction using one V_WMMA_F32_16X16X4_F32.
// A (16x4 f32, 2 VGPRs/lane): vgpr0 = acc, vgpr1 = 0.
//   lanes 0-15  -> A[m=lane, k=0] = acc ; k=1 = 0
//   lanes 16-31 -> A[m=lane-16, k=2] = acc ; k=3 = 0
// B = ones => D[m,n] = acc_m + acc_{m+16} = s_m (replicated over n).
// C/D layout: vgpr j holds D[j, lane] (lanes 0-15) and D[j+8, lane-16].
// So sum of 8 D vgprs gives s_0..7 on lanes 0-15 and s_8..15 on lanes 16-31;
// one xor-16 shuffle completes the 32-lane total (present in every lane).
__device__ __forceinline__ float wave_reduce_wmma(float acc) {
    v2f a; a[0] = acc;  a[1] = 0.0f;
    v2f b; b[0] = 1.0f; b[1] = 1.0f;
    v8f c = {};
    v8f d = __builtin_amdgcn_wmma_f32_16x16x4_f32(
        /*neg_a=*/false, a, /*neg_b=*/false, b,
        /*c_mod=*/(short)0, c, /*reuse_a=*/false, /*reuse_b=*/false);
    float s = ((d[0] + d[1]) + (d[2] + d[3])) + ((d[4] + d[5]) + (d[6] + d[7]));
    s += __shfl_xor(s, 16, 32);
    return s;
}

// Block reduction: WMMA wave reduce, then fixed-order sum across waves in LDS.
// Result valid in thread 0 only. Deterministic.
__device__ __forceinline__ float block_reduce(float acc, float* lds) {
    float w = wave_reduce_wmma(acc);
    int lane = threadIdx.x & 31;
    int wid  = threadIdx.x >> 5;
    if (lane == 0) lds[wid] = w;
    __syncthreads();
    float total = 0.0f;
    if (threadIdx.x == 0) {
        int nw = (int)(blockDim.x >> 5);
        for (int i = 0; i < nw; ++i) total += lds[i];
    }
    return total;
}

__global__ void __launch_bounds__(256)
bl_partial_kernel(const float* __restrict__ pred,
                  const float* __restrict__ tgt,
                  float* __restrict__ partials, int n4) {
    __shared__ float lds[8];
    const float4* p4 = reinterpret_cast<const float4*>(pred);
    const float4* t4 = reinterpret_cast<const float4*>(tgt);

    float acc = 0.0f;
    int stride = (int)(gridDim.x * blockDim.x);
    for (int j = (int)(blockIdx.x * blockDim.x + threadIdx.x); j < n4; j += stride) {
        float4 p = p4[j];           // global_load_b128 streaming
        float4 t = t4[j];
        acc += softplus_bce(p.x, t.x);
        acc += softplus_bce(p.y, t.y);
        acc += softplus_bce(p.z, t.z);
        acc += softplus_bce(p.w, t.w);
    }
    // All lanes reconverge here: EXEC is all-ones (WMMA requirement).
    float total = block_reduce(acc, lds);
    if (threadIdx.x == 0) partials[blockIdx.x] = total;
}

__global__ void __launch_bounds__(256)
bl_final_kernel(const float* __restrict__ partials, int nparts,
                const float* __restrict__ pred, const float* __restrict__ tgt,
                int tail_base, int n, float scale,
                float* __restrict__ out) {
    __shared__ float lds[8];
    float acc = 0.0f;
    // Fixed-order accumulation of block partials (deterministic).
    for (int j = (int)threadIdx.x; j < nparts; j += (int)blockDim.x)
        acc += partials[j];
    // Scalar tail (n % 4 != 0); empty for the 524288-element case.
    for (int j = tail_base + (int)threadIdx.x; j < n; j += (int)blockDim.x)
        acc += softplus_bce(pred[j], tgt[j]);

    float total = block_reduce(acc, lds);
    if (threadIdx.x == 0) out[0] = total * scale;
}

extern "C" void kernel_launch(void* const* d_in, const int* in_sizes, int n_in,
                              void* d_out, int out_size, void* d_ws, size_t ws_size,
                              hipStream_t stream) {
    const float* pred = (const float*)d_in[0];
    const float* tgt  = (const float*)d_in[1];
    float* out        = (float*)d_out;
    float* partials   = (float*)d_ws;

    const int n  = in_sizes[0];          // 8*1*256*256 = 524288
    const int n4 = n >> 2;               // float4 count
    const int tail_base = n4 << 2;

    const int threads = 256;
    int blocks = (n4 + threads - 1) / threads;   // 512 for the reference size
    if (blocks < 1)    blocks = 1;
    if (blocks > 2048) blocks = 2048;            // keep partials tiny & deterministic

    // final_weight == 1 + THETA == 6 exactly (see header comment); mean -> /n
    const float scale = (float)(6.0 / (double)n);

    bl_partial_kernel<<<blocks, threads, 0, stream>>>(pred, tgt, partials, n4);
    bl_final_kernel<<<1, threads, 0, stream>>>(partials, blocks,
                                               pred, tgt, tail_base, n, scale, out);
}